// GAT_17506286698901
// MI455X (gfx1250) — compile-verified
//
#include <hip/hip_runtime.h>
#include <math.h>

typedef __attribute__((ext_vector_type(2))) float v2f;
typedef __attribute__((ext_vector_type(8))) float v8f;

#define F_IN 64
#define HC   128   // HEADS * HID
#define NH   4     // heads
#define NEG_SLOPE 0.2f

// ---------------------------------------------------------------- init
__global__ void gat_init(float* m1, float* denom1, float* S1,
                         float* m2, float* denom2, float* S2, int N) {
  int i = blockIdx.x * blockDim.x + threadIdx.x;
  if (i < N * HC) S1[i] = 0.f;
  if (i < N * NH) { m1[i] = -INFINITY; denom1[i] = 0.f; }
  if (i < N)      { m2[i] = -INFINITY; denom2[i] = 0.f; S2[i] = 0.f; }
}

// ------------------------------------------------- WMMA fp32 GEMM (xl,xr)
// One wave per 16x16 output tile. A = x[16 x 64], B = W[64 x 128] col tile.
// A frag (32-bit 16x4): lane<16 -> K=k0,k0+1 ; lane>=16 -> K=k0+2,k0+3.
// B frag mirrors that striping over N lanes. C/D: 8 VGPRs, lanes16-31 = M+8.
__global__ void gat_gemm_wmma(const float* __restrict__ X,
                              const float* __restrict__ Wl,
                              const float* __restrict__ Wr,
                              float* __restrict__ XL,
                              float* __restrict__ XR, int N) {
  const float* W = blockIdx.z ? Wr : Wl;
  float* Out     = blockIdx.z ? XR : XL;
  const int lane = threadIdx.x;
  const int half = lane >> 4;        // 0: lanes 0-15, 1: lanes 16-31
  const int mn   = lane & 15;
  const int row0 = blockIdx.x * 16;
  const int col0 = blockIdx.y * 16;
  const int rowA = min(row0 + mn, N - 1);   // clamp for partial last tile

  v8f acc = {};
#pragma unroll
  for (int k0 = 0; k0 < F_IN; k0 += 4) {
    const int ka = k0 + 2 * half;
    v2f a, b;
    a.x = X[(size_t)rowA * F_IN + ka];
    a.y = X[(size_t)rowA * F_IN + ka + 1];
    b.x = W[(size_t)ka * HC + col0 + mn];
    b.y = W[(size_t)(ka + 1) * HC + col0 + mn];
    acc = __builtin_amdgcn_wmma_f32_16x16x4_f32(
        /*neg_a=*/false, a, /*neg_b=*/false, b,
        /*c_mod=*/(short)0, acc, /*reuse_a=*/false, /*reuse_b=*/false);
  }
#pragma unroll
  for (int r = 0; r < 8; ++r) {
    const int row = row0 + half * 8 + r;
    if (row < N) Out[(size_t)row * HC + col0 + mn] = acc[r];
  }
}

// --------------------------------------- layer-1 edge logits + segment max
// warp per edge; lane covers 4 channels; 8 lanes == one head.
__global__ void gat_edge1_logits(const long long* __restrict__ ei, int E, int EA,
                                 const float* __restrict__ XL,
                                 const float* __restrict__ XR,
                                 const float* __restrict__ att,
                                 float* __restrict__ logits,
                                 float* __restrict__ m1) {
  const int warp = (blockIdx.x * blockDim.x + threadIdx.x) >> 5;
  const int lane = threadIdx.x & 31;
  if (warp >= EA) return;
  int src, dst;
  if (warp < E) { src = (int)ei[warp]; dst = (int)ei[(size_t)E + warp]; }
  else          { src = dst = warp - E; }

  const int c0 = lane * 4;
  __builtin_prefetch(XL + (size_t)src * HC + c0, 0, 0);  // -> global_prefetch_b8
  const float4 xs = *(const float4*)(XL + (size_t)src * HC + c0);
  const float4 xd = *(const float4*)(XR + (size_t)dst * HC + c0);
  const float4 at = *(const float4*)(att + c0);

  float p = 0.f, v;
  v = xs.x + xd.x; p += at.x * (v > 0.f ? v : NEG_SLOPE * v);
  v = xs.y + xd.y; p += at.y * (v > 0.f ? v : NEG_SLOPE * v);
  v = xs.z + xd.z; p += at.z * (v > 0.f ? v : NEG_SLOPE * v);
  v = xs.w + xd.w; p += at.w * (v > 0.f ? v : NEG_SLOPE * v);

  p += __shfl_xor(p, 1);   // reduce within the 8 lanes of one head
  p += __shfl_xor(p, 2);
  p += __shfl_xor(p, 4);

  if ((lane & 7) == 0) {
    const int h = lane >> 3;
    logits[(size_t)warp * NH + h] = p;
    atomicMax(&m1[(size_t)dst * NH + h], p);
  }
}

// ------------------------- layer-1 fused alpha / denom / weighted aggregate
__global__ void gat_edge1_aggregate(const long long* __restrict__ ei, int E, int EA,
                                    const float* __restrict__ XL,
                                    const float* __restrict__ logits,
                                    const float* __restrict__ m1,
                                    float* __restrict__ denom1,
                                    float* __restrict__ S1) {
  const int warp = (blockIdx.x * blockDim.x + threadIdx.x) >> 5;
  const int lane = threadIdx.x & 31;
  if (warp >= EA) return;
  int src, dst;
  if (warp < E) { src = (int)ei[warp]; dst = (int)ei[(size_t)E + warp]; }
  else          { src = dst = warp - E; }

  const int h = lane >> 3;
  const float alpha =
      __expf(logits[(size_t)warp * NH + h] - m1[(size_t)dst * NH + h]);
  if ((lane & 7) == 0) atomicAdd(&denom1[(size_t)dst * NH + h], alpha);

  const int c0 = lane * 4;
  const float4 xs = *(const float4*)(XL + (size_t)src * HC + c0);
  float* Sp = S1 + (size_t)dst * HC + c0;
  atomicAdd(Sp + 0, alpha * xs.x);
  atomicAdd(Sp + 1, alpha * xs.y);
  atomicAdd(Sp + 2, alpha * xs.z);
  atomicAdd(Sp + 3, alpha * xs.w);
}

// ------------- node finalize: h = ELU(S/denom + b1); project hl, hr (warp/node)
__global__ void gat_node_finalize(const float* __restrict__ b1,
                                  const float* __restrict__ Wl2,
                                  const float* __restrict__ Wr2,
                                  const float* __restrict__ denom1,
                                  float* __restrict__ S1,     // in: sums, out: h
                                  float* __restrict__ hl,
                                  float* __restrict__ hr, int N) {
  const int warp = (blockIdx.x * blockDim.x + threadIdx.x) >> 5;
  const int lane = threadIdx.x & 31;
  if (warp >= N) return;
  const int c0 = lane * 4;
  const int h  = lane >> 3;
  const float inv = 1.f / denom1[(size_t)warp * NH + h];

  float4 s  = *(float4*)(S1 + (size_t)warp * HC + c0);
  const float4 bb = *(const float4*)(b1 + c0);
  const float4 wl = *(const float4*)(Wl2 + c0);
  const float4 wr = *(const float4*)(Wr2 + c0);

  float pl = 0.f, pr = 0.f, t, e;
  t = s.x * inv + bb.x; e = t > 0.f ? t : __expf(t) - 1.f; s.x = e; pl += e * wl.x; pr += e * wr.x;
  t = s.y * inv + bb.y; e = t > 0.f ? t : __expf(t) - 1.f; s.y = e; pl += e * wl.y; pr += e * wr.y;
  t = s.z * inv + bb.z; e = t > 0.f ? t : __expf(t) - 1.f; s.z = e; pl += e * wl.z; pr += e * wr.z;
  t = s.w * inv + bb.w; e = t > 0.f ? t : __expf(t) - 1.f; s.w = e; pl += e * wl.w; pr += e * wr.w;
  *(float4*)(S1 + (size_t)warp * HC + c0) = s;

  for (int msk = 1; msk < 32; msk <<= 1) {
    pl += __shfl_xor(pl, msk);
    pr += __shfl_xor(pr, msk);
  }
  if (lane == 0) { hl[warp] = pl; hr[warp] = pr; }
}

// ------------------------------------------- layer-2 (scalar) edge passes
__global__ void gat_edge2_logits(const long long* __restrict__ ei, int E, int EA,
                                 const float* __restrict__ hl,
                                 const float* __restrict__ hr,
                                 const float* __restrict__ att2,
                                 float* __restrict__ logits2,
                                 float* __restrict__ m2) {
  const int e = blockIdx.x * blockDim.x + threadIdx.x;
  if (e >= EA) return;
  int src, dst;
  if (e < E) { src = (int)ei[e]; dst = (int)ei[(size_t)E + e]; }
  else       { src = dst = e - E; }
  const float v = hl[src] + hr[dst];
  const float lr = v > 0.f ? v : NEG_SLOPE * v;
  const float logit = att2[0] * lr;
  logits2[e] = logit;
  atomicMax(&m2[dst], logit);
}

__global__ void gat_edge2_aggregate(const long long* __restrict__ ei, int E, int EA,
                                    const float* __restrict__ hl,
                                    const float* __restrict__ logits2,
                                    const float* __restrict__ m2,
                                    float* __restrict__ denom2,
                                    float* __restrict__ S2) {
  const int e = blockIdx.x * blockDim.x + threadIdx.x;
  if (e >= EA) return;
  int src, dst;
  if (e < E) { src = (int)ei[e]; dst = (int)ei[(size_t)E + e]; }
  else       { src = dst = e - E; }
  const float alpha = __expf(logits2[e] - m2[dst]);
  atomicAdd(&denom2[dst], alpha);
  atomicAdd(&S2[dst], alpha * hl[src]);
}

__global__ void gat_final(const float* __restrict__ S2,
                          const float* __restrict__ denom2,
                          const float* __restrict__ b2,
                          float* __restrict__ out, int N) {
  const int n = blockIdx.x * blockDim.x + threadIdx.x;
  if (n < N) out[n] = S2[n] / denom2[n] + b2[0];
}

// ---------------------------------------------------------------- launch
extern "C" void kernel_launch(void* const* d_in, const int* in_sizes, int n_in,
                              void* d_out, int out_size, void* d_ws, size_t ws_size,
                              hipStream_t stream) {
  const float*     x    = (const float*)d_in[0];
  const long long* ei   = (const long long*)d_in[1];   // int64 edge_index [2,E]
  const float*     Wl1  = (const float*)d_in[2];
  const float*     Wr1  = (const float*)d_in[3];
  const float*     att1 = (const float*)d_in[4];
  const float*     b1   = (const float*)d_in[5];
  const float*     Wl2  = (const float*)d_in[6];
  const float*     Wr2  = (const float*)d_in[7];
  const float*     att2 = (const float*)d_in[8];
  const float*     b2   = (const float*)d_in[9];
  float* out = (float*)d_out;

  const int N  = in_sizes[0] / F_IN;
  const int E  = in_sizes[1] / 2;
  const int EA = E + N;                 // + self loops

  // workspace layout (floats)
  float* ws      = (float*)d_ws;
  float* XL      = ws;                       ws += (size_t)N * HC;
  float* XR      = ws;                       ws += (size_t)N * HC;
  float* S1      = ws;                       ws += (size_t)N * HC;  // sums, then h
  float* m1      = ws;                       ws += (size_t)N * NH;
  float* denom1  = ws;                       ws += (size_t)N * NH;
  float* logits1 = ws;                       ws += (size_t)EA * NH;
  float* hl      = ws;                       ws += N;
  float* hr      = ws;                       ws += N;
  float* m2      = ws;                       ws += N;
  float* denom2  = ws;                       ws += N;
  float* S2      = ws;                       ws += N;
  float* logits2 = ws;                       ws += EA;
  (void)ws_size; (void)n_in; (void)out_size;

  // 1. init
  gat_init<<<(N * HC + 255) / 256, 256, 0, stream>>>(m1, denom1, S1, m2, denom2, S2, N);

  // 2. xl = x@Wl1, xr = x@Wr1 via v_wmma_f32_16x16x4_f32
  dim3 ggrid((N + 15) / 16, HC / 16, 2);
  gat_gemm_wmma<<<ggrid, 32, 0, stream>>>(x, Wl1, Wr1, XL, XR, N);

  // 3. layer-1 edge logits + segment max (warp per edge)
  const int wpb = 256 / 32;
  gat_edge1_logits<<<(EA + wpb - 1) / wpb, 256, 0, stream>>>(ei, E, EA, XL, XR, att1,
                                                             logits1, m1);
  // 4. layer-1 fused softmax-weight aggregation
  gat_edge1_aggregate<<<(EA + wpb - 1) / wpb, 256, 0, stream>>>(ei, E, EA, XL, logits1,
                                                                m1, denom1, S1);
  // 5. node finalize + layer-2 projections (warp per node)
  gat_node_finalize<<<(N + wpb - 1) / wpb, 256, 0, stream>>>(b1, Wl2, Wr2, denom1, S1,
                                                             hl, hr, N);
  // 6-7. layer-2 scalar edge passes (thread per edge)
  gat_edge2_logits<<<(EA + 255) / 256, 256, 0, stream>>>(ei, E, EA, hl, hr, att2,
                                                         logits2, m2);
  gat_edge2_aggregate<<<(EA + 255) / 256, 256, 0, stream>>>(ei, E, EA, hl, logits2,
                                                            m2, denom2, S2);
  // 8. output
  gat_final<<<(N + 255) / 256, 256, 0, stream>>>(S2, denom2, b2, out, N);
}